// GAT_1838246003175
// MI455X (gfx1250) — compile-verified
//
#include <hip/hip_runtime.h>

// ---------------------------------------------------------------------------
// GAT 2-layer forward for MI455X (gfx1250).
// GEMMs via V_WMMA_F32_16X16X4_F32 (fp32 fidelity; total GEMM work ~1.3 GFLOP
// is negligible). Edge phase = L2-resident gather + scatter atomics, the real
// ~1.4 GB of traffic (~60 us floor at 23.3 TB/s).
// ---------------------------------------------------------------------------

static constexpr int NN     = 50000;
static constexpr int NEDGE0 = 800000;            // raw edges
static constexpr int NEDGE  = 850000;            // + self loops
static constexpr int CCH    = 64;                // channels per head

typedef float v2f __attribute__((ext_vector_type(2)));
typedef float v8f __attribute__((ext_vector_type(8)));

// ---- order-preserving float<->uint for atomic max ----
__device__ __forceinline__ unsigned enc_ord(float f) {
  unsigned u = __float_as_uint(f);
  return (u & 0x80000000u) ? ~u : (u | 0x80000000u);
}
__device__ __forceinline__ float dec_ord(unsigned u) {
  unsigned v = (u & 0x80000000u) ? (u & 0x7fffffffu) : ~u;
  return __uint_as_float(v);
}
static constexpr unsigned ENC_NEG_INF = 0x007FFFFFu; // enc_ord(-inf)

__device__ __forceinline__ void edge_ep(int e, const long long* __restrict__ ei,
                                        int& s, int& d) {
  if (e < NEDGE0) { s = (int)ei[e]; d = (int)ei[(long long)NEDGE0 + e]; }
  else            { s = d = e - NEDGE0; }          // self loop
}

// ---------------------------- utility kernels ------------------------------
__global__ void kzero_f32(float* __restrict__ p, long long n) {
  long long i = (long long)blockIdx.x * blockDim.x + threadIdx.x;
  if (i < n) p[i] = 0.f;
}
__global__ void kfill_u32(unsigned* __restrict__ p, unsigned v, long long n) {
  long long i = (long long)blockIdx.x * blockDim.x + threadIdx.x;
  if (i < n) p[i] = v;
}

// ------------------------- fp32 WMMA GEMM ----------------------------------
// C[M,Nn] = A[M,K] * B[K,Nn].  One 16x16 output tile per wave32.
// A 16x4 frag (ISA 7.12.2): lane(0..15)=M; hi half-wave holds K+2; the two
// K-consecutive elements per lane are contiguous in memory -> one b64 load.
// B frag: VGPR r -> rows {k, k+2} striped across lane halves.
// C/D: v8f, row = r + 8*hi, col = lane&15.
// K is a template constant: main loop unguarded + unrolled, tail branchless.
template <int K>
__global__ void gemm_wmma_f32(const float* __restrict__ A,
                              const float* __restrict__ B,
                              float* __restrict__ Cout,
                              int M, int Nn) {
  int wave = (int)((blockIdx.x * (long long)blockDim.x + threadIdx.x) >> 5);
  int lane = threadIdx.x & 31;
  int ntiles = Nn >> 4;
  if (wave >= (M >> 4) * ntiles) return;
  int mt = wave / ntiles, nt = wave % ntiles;
  int lo = lane & 15, hi = lane >> 4;
  int m  = (mt << 4) + lo;                        // A row for this lane
  int nn = (nt << 4) + lo;                        // B/C column for this lane
  const float* __restrict__ Arow = A + (long long)m * K;
  v8f acc = {};
  constexpr int KM = K & ~3;                      // multiple-of-4 prefix
#pragma unroll
  for (int kb = 0; kb < KM; kb += 4) {
    int k0 = kb + 2 * hi;
    v2f av = *(const v2f*)(Arow + k0);            // contiguous 8B load
    v2f bv = { B[(long long)k0 * Nn + nn],
               B[(long long)(k0 + 1) * Nn + nn] };
    acc = __builtin_amdgcn_wmma_f32_16x16x4_f32(
        false, av, false, bv, (short)0, acc, false, false);
  }
  if constexpr ((K & 3) != 0) {                   // branchless K tail
    int k0 = KM + 2 * hi;
    bool q0 = k0 < K, q1 = (k0 + 1) < K;
    float a0 = Arow[q0 ? k0 : 0];                  a0 = q0 ? a0 : 0.f;
    float a1 = Arow[q1 ? (k0 + 1) : 0];            a1 = q1 ? a1 : 0.f;
    float b0 = B[(long long)(q0 ? k0 : 0) * Nn + nn];        b0 = q0 ? b0 : 0.f;
    float b1 = B[(long long)(q1 ? (k0 + 1) : 0) * Nn + nn];  b1 = q1 ? b1 : 0.f;
    v2f av = {a0, a1};
    v2f bv = {b0, b1};
    acc = __builtin_amdgcn_wmma_f32_16x16x4_f32(
        false, av, false, bv, (short)0, acc, false, false);
  }
#pragma unroll
  for (int r = 0; r < 8; ++r) {
    int row = (mt << 4) + r + 8 * hi;
    Cout[(long long)row * Nn + nn] = acc[r];
  }
}

// ---------------------- per-node attention dots ----------------------------
template <int H>
__global__ void node_att(const float* __restrict__ h,
                         const float* __restrict__ att_s,
                         const float* __restrict__ att_d,
                         float* __restrict__ asv, float* __restrict__ adv) {
  int i = blockIdx.x * blockDim.x + threadIdx.x;
  if (i >= NN * H) return;
  int node = i / H, head = i % H;
  const float* hp = h + (long long)node * H * CCH + head * CCH;
  const float* sp = att_s + head * CCH;
  const float* dp = att_d + head * CCH;
  float s = 0.f, d = 0.f;
#pragma unroll 8
  for (int c = 0; c < CCH; ++c) { float v = hp[c]; s += v * sp[c]; d += v * dp[c]; }
  asv[i] = s;
  adv[i] = d;
}

// ------------------------- edge softmax phases -----------------------------
template <int H>
__global__ void edge_logit(const long long* __restrict__ ei,
                           const float* __restrict__ asv,
                           const float* __restrict__ adv,
                           float* __restrict__ elog,
                           unsigned* __restrict__ menc) {
  long long i = (long long)blockIdx.x * blockDim.x + threadIdx.x;
  if (i >= (long long)NEDGE * H) return;
  int e = (int)(i / H), head = (int)(i % H);
  int s, d; edge_ep(e, ei, s, d);
  float lg = asv[s * H + head] + adv[d * H + head];
  lg = lg > 0.f ? lg : 0.2f * lg;                 // leaky_relu(0.2)
  elog[i] = lg;
  atomicMax(&menc[d * H + head], enc_ord(lg));
}

template <int H>
__global__ void edge_exp(const long long* __restrict__ ei,
                         float* __restrict__ elog,
                         const unsigned* __restrict__ menc,
                         float* __restrict__ den) {
  long long i = (long long)blockIdx.x * blockDim.x + threadIdx.x;
  if (i >= (long long)NEDGE * H) return;
  int e = (int)(i / H), head = (int)(i % H);
  int s, d; edge_ep(e, ei, s, d);
  (void)s;
  float m = dec_ord(menc[d * H + head]);
  unsigned mb = __float_as_uint(m);
  if ((mb & 0x7f800000u) == 0x7f800000u) m = 0.f;  // isfinite fixup
  float ex = __expf(elog[i] - m);
  elog[i] = ex;
  atomicAdd(&den[d * H + head], ex);
}

// msg = alpha * h[src]; out[dst] += msg.  One thread per (edge, 4 channels);
// a wave covers contiguous float4 chunks -> fully coalesced gathers.
template <int H>
__global__ void edge_scatter(const long long* __restrict__ ei,
                             const float* __restrict__ elog,
                             const float* __restrict__ den,
                             const float* __restrict__ h,
                             float* __restrict__ agg) {
  constexpr int CHUNKS = H * CCH / 4;             // float4 chunks per edge (16/32)
  long long i = (long long)blockIdx.x * blockDim.x + threadIdx.x;
  if (i >= (long long)NEDGE * CHUNKS) return;
  int e = (int)(i / CHUNKS);
  int r = (int)(i % CHUNKS);
  int c4 = r * 4;
  int head = c4 / CCH;
  int s, d; edge_ep(e, ei, s, d);
  float alpha = elog[(long long)e * H + head] / (den[d * H + head] + 1e-16f);
  const float4 hv = *(const float4*)(h + (long long)s * H * CCH + c4);
  float* ap = agg + (long long)d * H * CCH + c4;
  atomicAdd(ap + 0, alpha * hv.x);
  atomicAdd(ap + 1, alpha * hv.y);
  atomicAdd(ap + 2, alpha * hv.z);
  atomicAdd(ap + 3, alpha * hv.w);
}

// --------------------------- epilogues -------------------------------------
__global__ void bias_relu(float* __restrict__ p, const float* __restrict__ b,
                          int HC, long long n) {
  long long i = (long long)blockIdx.x * blockDim.x + threadIdx.x;
  if (i >= n) return;
  float v = p[i] + b[i % HC];
  p[i] = v > 0.f ? v : 0.f;
}
__global__ void bias_add(float* __restrict__ p, const float* __restrict__ b,
                         int HC, long long n) {
  long long i = (long long)blockIdx.x * blockDim.x + threadIdx.x;
  if (i >= n) return;
  p[i] += b[i % HC];
}

// ---------------------------------------------------------------------------
extern "C" void kernel_launch(void* const* d_in, const int* in_sizes, int n_in,
                              void* d_out, int out_size, void* d_ws, size_t ws_size,
                              hipStream_t stream) {
  (void)in_sizes; (void)n_in; (void)out_size; (void)ws_size;
  const float*     x   = (const float*)d_in[0];
  const long long* ei  = (const long long*)d_in[1];  // int64 [2, 800000]
  const float*     W1  = (const float*)d_in[2];
  const float*     as1 = (const float*)d_in[3];
  const float*     ad1 = (const float*)d_in[4];
  const float*     b1  = (const float*)d_in[5];
  const float*     W2  = (const float*)d_in[6];
  const float*     as2 = (const float*)d_in[7];
  const float*     ad2 = (const float*)d_in[8];
  const float*     b2  = (const float*)d_in[9];
  float* out = (float*)d_out;

  // workspace layout (all fp32-sized slots)
  float* ws = (float*)d_ws;
  float*    h1   = ws;              ws += (long long)NN * 128;
  float*    agg1 = ws;              ws += (long long)NN * 128;
  float*    h2   = ws;              ws += (long long)NN * 64;
  float*    as1v = ws;              ws += NN * 2;
  float*    ad1v = ws;              ws += NN * 2;
  unsigned* m1   = (unsigned*)ws;   ws += NN * 2;
  float*    den1 = ws;              ws += NN * 2;
  float*    as2v = ws;              ws += NN;
  float*    ad2v = ws;              ws += NN;
  unsigned* m2   = (unsigned*)ws;   ws += NN;
  float*    den2 = ws;              ws += NN;
  float*    elog = ws;              ws += (long long)NEDGE * 2;

  const int T = 256;
  auto blk = [](long long n, int t) { return (unsigned)((n + t - 1) / t); };

  // init accumulators (harness poisons ws/out with 0xAA)
  kzero_f32<<<blk((long long)NN * 128, T), T, 0, stream>>>(agg1, (long long)NN * 128);
  kzero_f32<<<blk(NN * 2, T), T, 0, stream>>>(den1, NN * 2);
  kzero_f32<<<blk(NN, T), T, 0, stream>>>(den2, NN);
  kzero_f32<<<blk((long long)NN * 64, T), T, 0, stream>>>(out, (long long)NN * 64);
  kfill_u32<<<blk(NN * 2, T), T, 0, stream>>>(m1, ENC_NEG_INF, NN * 2);
  kfill_u32<<<blk(NN, T), T, 0, stream>>>(m2, ENC_NEG_INF, NN);

  // ---- layer 1: heads=2, concat=True (out 128) ----
  {
    long long tiles = (long long)(NN / 16) * (128 / 16);
    gemm_wmma_f32<34><<<blk(tiles * 32, T), T, 0, stream>>>(x, W1, h1, NN, 128);
  }
  node_att<2><<<blk(NN * 2, T), T, 0, stream>>>(h1, as1, ad1, as1v, ad1v);
  edge_logit<2><<<blk((long long)NEDGE * 2, T), T, 0, stream>>>(ei, as1v, ad1v, elog, m1);
  edge_exp<2><<<blk((long long)NEDGE * 2, T), T, 0, stream>>>(ei, elog, m1, den1);
  edge_scatter<2><<<blk((long long)NEDGE * 32, T), T, 0, stream>>>(ei, elog, den1, h1, agg1);
  bias_relu<<<blk((long long)NN * 128, T), T, 0, stream>>>(agg1, b1, 128, (long long)NN * 128);

  // ---- layer 2: heads=1, concat=False (mean over 1 head == identity) ----
  {
    long long tiles = (long long)(NN / 16) * (64 / 16);
    gemm_wmma_f32<128><<<blk(tiles * 32, T), T, 0, stream>>>(agg1, W2, h2, NN, 64);
  }
  node_att<1><<<blk(NN, T), T, 0, stream>>>(h2, as2, ad2, as2v, ad2v);
  edge_logit<1><<<blk(NEDGE, T), T, 0, stream>>>(ei, as2v, ad2v, elog, m2);
  edge_exp<1><<<blk(NEDGE, T), T, 0, stream>>>(ei, elog, m2, den2);
  edge_scatter<1><<<blk((long long)NEDGE * 16, T), T, 0, stream>>>(ei, elog, den2, h2, out);
  bias_add<<<blk((long long)NN * 64, T), T, 0, stream>>>(out, b2, 64, (long long)NN * 64);
}